// Turn_map_into_waves_40570261078379
// MI455X (gfx1250) — compile-verified
//
#include <hip/hip_runtime.h>
#include <hip/hip_bf16.h>

// out[b,d] = mean_i attn[b, i, i+d]  (superdiagonal means), B=32, N=2048.
// Skewed view A'[i,d] = A[i,i+d]: diagonal sums == column sums of A'.
// Column sums via V_WMMA_F32_16X16X4_F32: D = ones(16x4) x B(4x16) + C.
// One 32-thread (1 wave) block per 128-row chunk of one 16-wide diagonal band.

typedef __attribute__((ext_vector_type(2))) float v2f;
typedef __attribute__((ext_vector_type(8))) float v8f;

#define NN 2048
#define NB 32
#define JBLK 128        // number of 16-wide diagonal bands
#define CHUNK_ROWS 128  // rows per wave-task
#define NCHUNK 16       // 2048 / 128
#define RS (NN + 1)     // element stride of the skewed diagonal walk

__global__ __launch_bounds__(32)
void diag_sum_wmma(const float* __restrict__ attn,
                   float* __restrict__ partial,   // [NB][JBLK][NCHUNK][16] (det path)
                   float* __restrict__ outAtomic, // [NB][NN]              (atomic path)
                   int useAtomic)
{
    const int chunk   = blockIdx.x;   // 0..15  (row chunk along the band)
    const int J       = blockIdx.y;   // 0..127 (band: d in [16J, 16J+15])
    const int b       = blockIdx.z;   // 0..31
    const int colbase = J << 4;
    const int rowbase = chunk * CHUNK_ROWS;
    if (rowbase >= NN - colbase) return;          // band shorter than this chunk

    const int lane = threadIdx.x;  // wave32
    const int c    = lane & 15;    // column within band (d = colbase + c)
    const int hi   = lane >> 4;

    const float* __restrict__ base = attn + (size_t)b * NN * NN;
    const int i0   = rowbase + hi;                 // first row this lane touches
    const int idx0 = i0 * RS + colbase + c;        // element index A[i0, i0+colbase+c]

    v8f acc0 = {0.f,0.f,0.f,0.f,0.f,0.f,0.f,0.f};
    v8f acc1 = {0.f,0.f,0.f,0.f,0.f,0.f,0.f,0.f};
    v2f a1;  a1.x = 1.0f;  a1.y = 1.0f;            // A = all-ones (16x4)

    if (rowbase + colbase + 142 < NN) {
        // ---- fast path: every load in-bounds, no predication ----
        const float* __restrict__ p = base + idx0;
        #pragma unroll 4
        for (int step = 0; step < CHUNK_ROWS / 8; ++step) {
            v2f b0, b1;
            b0.x = p[0];        // rows 8*step + {0,1}
            b0.y = p[2 * RS];   // rows 8*step + {2,3}
            b1.x = p[4 * RS];   // rows 8*step + {4,5}
            b1.y = p[6 * RS];   // rows 8*step + {6,7}
            p += 8 * RS;
            acc0 = __builtin_amdgcn_wmma_f32_16x16x4_f32(false, a1, false, b0,
                                                         (short)0, acc0, false, false);
            acc1 = __builtin_amdgcn_wmma_f32_16x16x4_f32(false, a1, false, b1,
                                                         (short)0, acc1, false, false);
        }
    } else {
        // ---- tail path (last chunk of the band): branchless clamp + select ----
        int idx = idx0;
        int j   = i0 + colbase + c;                // global column for v=0
        const int maxidx = NN * NN - 1;            // keep loads inside this batch
        #pragma unroll 2
        for (int step = 0; step < CHUNK_ROWS / 4; ++step) {
            const float x0 = base[min(idx,          maxidx)];
            const float x1 = base[min(idx + 2 * RS, maxidx)];
            v2f bm;
            bm.x = (j     < NN) ? x0 : 0.0f;
            bm.y = (j + 2 < NN) ? x1 : 0.0f;
            idx += 4 * RS;
            j   += 4;
            acc0 = __builtin_amdgcn_wmma_f32_16x16x4_f32(false, a1, false, bm,
                                                         (short)0, acc0, false, false);
        }
    }

    // D layout: VGPR0, lanes 0..15 hold D[0][n] = column sums.
    const float colsum = acc0[0] + acc1[0];
    if (lane < 16) {
        if (useAtomic) {
            atomicAdd(&outAtomic[(size_t)b * NN + colbase + lane], colsum);
        } else {
            partial[(((size_t)b * JBLK + J) * NCHUNK + chunk) * 16 + lane] = colsum;
        }
    }
}

__global__ void finalize_det(const float* __restrict__ partial, float* __restrict__ out)
{
    const int idx = blockIdx.x * blockDim.x + threadIdx.x;
    if (idx >= NB * NN) return;
    const int b = idx / NN;
    const int d = idx % NN;
    const int J = d >> 4;
    const int n = d & 15;
    const int nrows   = NN - (J << 4);
    const int nchunks = (nrows + CHUNK_ROWS - 1) / CHUNK_ROWS;
    const float* p = partial + (((size_t)b * JBLK + J) * NCHUNK) * 16 + n;
    float s = 0.0f;
    for (int cix = 0; cix < nchunks; ++cix) s += p[cix * 16];
    out[idx] = s / (float)(NN - d);
}

__global__ void finalize_div(float* __restrict__ out)
{
    const int idx = blockIdx.x * blockDim.x + threadIdx.x;
    if (idx >= NB * NN) return;
    const int d = idx % NN;
    out[idx] = out[idx] / (float)(NN - d);
}

extern "C" void kernel_launch(void* const* d_in, const int* in_sizes, int n_in,
                              void* d_out, int out_size, void* d_ws, size_t ws_size,
                              hipStream_t stream)
{
    (void)in_sizes; (void)n_in; (void)out_size;
    const float* attn = (const float*)d_in[0];
    float* out = (float*)d_out;

    const size_t need = (size_t)NB * JBLK * NCHUNK * 16 * sizeof(float);  // 4 MB
    const dim3 grid(NCHUNK, JBLK, NB);
    const int fin_blocks = (NB * NN + 255) / 256;

    if (ws_size >= need) {
        // Deterministic: per-chunk partials + serial finalize (no atomics).
        float* partial = (float*)d_ws;
        diag_sum_wmma<<<grid, 32, 0, stream>>>(attn, partial, nullptr, 0);
        finalize_det<<<fin_blocks, 256, 0, stream>>>(partial, out);
    } else {
        // Fallback: f32 atomics straight into d_out.
        hipMemsetAsync(out, 0, (size_t)NB * NN * sizeof(float), stream);
        diag_sum_wmma<<<grid, 32, 0, stream>>>(attn, nullptr, out, 1);
        finalize_div<<<fin_blocks, 256, 0, stream>>>(out);
    }
}